// GNN_60387240182014
// MI455X (gfx1250) — compile-verified
//
#include <hip/hip_runtime.h>

#define N_NODES  300000
#define N_EDGES  600000
#define N_GRAPHS 12000
#define HIDDEN   128
#define F_ATOM   9
#define V_ATOM   119
#define D_ATOM   9

typedef __attribute__((ext_vector_type(2))) float v2f;
typedef __attribute__((ext_vector_type(8))) float v8f;

// ---------------------------------------------------------------- utilities
__global__ void zero_f_kernel(float* p, int n) {
    int i = blockIdx.x * blockDim.x + threadIdx.x;
    if (i < n) p[i] = 0.0f;
}

// deg[dst] += 1 per edge
__global__ void deg_kernel(const int* dst, float* deg, int n) {
    int e = blockIdx.x * blockDim.x + threadIdx.x;
    if (e < n) atomicAdd(&deg[dst[e]], 1.0f);
}

// in-place: deg -> rsqrt(deg + 1)
__global__ void dinv_kernel(float* deg, int n) {
    int i = blockIdx.x * blockDim.x + threadIdx.x;
    if (i < n) deg[i] = rsqrtf(deg[i] + 1.0f);
}

// ---------------------------------------------------------------- embedding
// h0[n][d] = sum_f atom_emb[f][x[n][f]][d]
__global__ void embed_kernel(const int* __restrict__ x,
                             const float* __restrict__ emb,
                             float* __restrict__ h0) {
    int n = blockIdx.x * blockDim.x + threadIdx.x;
    if (n >= N_NODES) return;
    float acc[D_ATOM];
#pragma unroll
    for (int d = 0; d < D_ATOM; ++d) acc[d] = 0.0f;
#pragma unroll
    for (int f = 0; f < F_ATOM; ++f) {
        int xi = x[n * F_ATOM + f];
        const float* t = emb + ((size_t)f * V_ATOM + xi) * D_ATOM;
#pragma unroll
        for (int d = 0; d < D_ATOM; ++d) acc[d] += t[d];
    }
#pragma unroll
    for (int d = 0; d < D_ATOM; ++d) h0[(size_t)n * D_ATOM + d] = acc[d];
}

// ---------------------------------------------------------------- skinny matmul (K=9)
// out[n][j] = sum_d h0[n][d] * W1[d][j]
__global__ void mm_k9_kernel(const float* __restrict__ h0,
                             const float* __restrict__ W1,
                             float* __restrict__ out) {
    int tid = blockIdx.x * blockDim.x + threadIdx.x; // n*128 + j
    int n = tid >> 7, j = tid & 127;
    if (n >= N_NODES) return;
    float s = 0.0f;
#pragma unroll
    for (int d = 0; d < D_ATOM; ++d)
        s += h0[(size_t)n * D_ATOM + d] * W1[d * HIDDEN + j];
    out[tid] = s;
}

// ---------------------------------------------------------------- GCN self term + bias
// out[n][j] = hw[n][j]*dinv[n]^2 + b[j]
__global__ void self_kernel(const float* __restrict__ hw,
                            const float* __restrict__ dinv,
                            const float* __restrict__ b,
                            float* __restrict__ out) {
    int tid = blockIdx.x * blockDim.x + threadIdx.x;
    int n = tid >> 7, j = tid & 127;
    if (n >= N_NODES) return;
    float di = dinv[n];
    out[tid] = hw[tid] * di * di + b[j];
}

// ---------------------------------------------------------------- edge scatter-add
// out[dst][j] += hw[src][j] * dinv[src]*dinv[dst]
__global__ void scatter_kernel(const int* __restrict__ ei,
                               const float* __restrict__ hw,
                               const float* __restrict__ dinv,
                               float* __restrict__ out) {
    int tid = blockIdx.x * blockDim.x + threadIdx.x; // e*128 + j
    int e = tid >> 7, j = tid & 127;
    if (e >= N_EDGES) return;
    int s = ei[e];
    int d = ei[N_EDGES + e];
    float coef = dinv[s] * dinv[d];
    atomicAdd(&out[(size_t)d * HIDDEN + j], hw[(size_t)s * HIDDEN + j] * coef);
}

// ---------------------------------------------------------------- WMMA GEMM
// D = (applyRelu ? relu(A) : A) @ B + bias, A: M x128, B: 128x128, M % 16 == 0.
// 256 threads = 8 waves per block; block handles a 16-row tile, wave w does col tile w.
// Uses V_WMMA_F32_16X16X4_F32, accumulator chained across K=128 in steps of 4.
__global__ void gemm_wmma_kernel(const float* __restrict__ A,
                                 const float* __restrict__ B,
                                 const float* __restrict__ bias,
                                 float* __restrict__ D,
                                 int M, int applyRelu) {
    __shared__ float As[16 * HIDDEN]; // 8 KB A tile, shared by the 8 waves
    int tileM = blockIdx.x;
    const float* Arow = A + (size_t)tileM * 16 * HIDDEN;
    for (int i = threadIdx.x; i < 16 * HIDDEN; i += 256) {
        float v = Arow[i];
        As[i] = applyRelu ? fmaxf(v, 0.0f) : v;
    }
    __syncthreads();

    int wave = threadIdx.x >> 5;   // 0..7 -> N tile
    int lane = threadIdx.x & 31;
    int half = lane >> 4;          // 0: K pair {0,1}, 1: K pair {2,3}
    int l16  = lane & 15;          // A row / B col / D col within tile
    int tileN = wave * 16;
    int kk = half * 2;

    v8f c = {};
    for (int k = 0; k < HIDDEN; k += 4) {
        v2f a, b;
        // A 16x4 fragment: lanes 0-15 hold M=0..15 K={k,k+1}; lanes 16-31 K={k+2,k+3}
        a.x = As[l16 * HIDDEN + k + kk];
        a.y = As[l16 * HIDDEN + k + kk + 1];
        // B 4x16 fragment: lanes 0-15 hold N=0..15 of rows K={k,k+1}; lanes 16-31 rows K={k+2,k+3}
        b.x = B[(k + kk) * HIDDEN + tileN + l16];
        b.y = B[(k + kk + 1) * HIDDEN + tileN + l16];
        c = __builtin_amdgcn_wmma_f32_16x16x4_f32(
                /*neg_a=*/false, a, /*neg_b=*/false, b,
                /*c_mod=*/(short)0, c, /*reuse_a=*/false, /*reuse_b=*/false);
    }

    // D 16x16: lane<16 vgpr r -> M=r; lane>=16 vgpr r -> M=8+r; N = l16
    int rowBase = tileM * 16 + half * 8;
    float bv = bias ? bias[tileN + l16] : 0.0f;
#pragma unroll
    for (int r = 0; r < 8; ++r) {
        D[(size_t)(rowBase + r) * HIDDEN + tileN + l16] = c[r] + bv;
    }
}

// ---------------------------------------------------------------- pooling
__global__ void pool_kernel(const int* __restrict__ batch,
                            const float* __restrict__ h,
                            float* __restrict__ gs,
                            float* __restrict__ cnt) {
    int tid = blockIdx.x * blockDim.x + threadIdx.x; // n*128 + j
    int n = tid >> 7, j = tid & 127;
    if (n >= N_NODES) return;
    int g = batch[n];
    atomicAdd(&gs[(size_t)g * HIDDEN + j], h[tid]);
    if (j == 0) atomicAdd(&cnt[g], 1.0f);
}

__global__ void mean_kernel(float* gs, const float* __restrict__ cnt) {
    int tid = blockIdx.x * blockDim.x + threadIdx.x;
    int g = tid >> 7;
    if (g >= N_GRAPHS) return;
    gs[tid] /= fmaxf(cnt[g], 1.0f);
}

// ---------------------------------------------------------------- launch
extern "C" void kernel_launch(void* const* d_in, const int* in_sizes, int n_in,
                              void* d_out, int out_size, void* d_ws, size_t ws_size,
                              hipStream_t stream) {
    (void)in_sizes; (void)n_in; (void)out_size; (void)ws_size;

    const int*   x        = (const int*)  d_in[0];   // (300000, 9)
    const int*   ei       = (const int*)  d_in[1];   // (2, 600000)
    // d_in[2] edge_attr: unused by reference output
    const int*   batch    = (const int*)  d_in[3];   // (300000,)
    const float* atom_emb = (const float*)d_in[4];   // (9, 119, 9)
    // d_in[5] bond_emb: unused
    const float* W1  = (const float*)d_in[6];        // (9, 128)
    const float* b1  = (const float*)d_in[7];        // (128,)
    const float* W2  = (const float*)d_in[8];        // (128, 128)
    const float* b2  = (const float*)d_in[9];        // (128,)
    const float* Wfc = (const float*)d_in[10];       // (128, 128)
    const float* bfc = (const float*)d_in[11];       // (128,)
    float* out = (float*)d_out;                      // (12000, 128)

    // Workspace layout (floats), ~319 MB total:
    float* ws = (float*)d_ws;
    size_t off = 0;
    float* dinv = ws + off; off += 300032;                 // deg -> dinv (in place)
    float* h0   = ws + off; off += 2700032;                // (300000, 9); dead after mm1
    float* bufA = ws + off; off += (size_t)N_NODES * HIDDEN; // hW1, later hW2
    float* bufB = ws + off; off += (size_t)N_NODES * HIDDEN; // h1, later h2
    float* gs   = h0;                                      // reuse h0 region: (12000,128)
    float* cnt  = h0 + (size_t)N_GRAPHS * HIDDEN;          // (12000,)

    const int T = 256;
    const int NV  = N_NODES * HIDDEN;                      // 38.4M
    const int EV  = N_EDGES * HIDDEN;                      // 76.8M

    // degrees -> dinv
    zero_f_kernel<<<(N_NODES + T - 1) / T, T, 0, stream>>>(dinv, N_NODES);
    deg_kernel   <<<(N_EDGES + T - 1) / T, T, 0, stream>>>(ei + N_EDGES, dinv, N_EDGES);
    dinv_kernel  <<<(N_NODES + T - 1) / T, T, 0, stream>>>(dinv, N_NODES);

    // embedding
    embed_kernel<<<(N_NODES + T - 1) / T, T, 0, stream>>>(x, atom_emb, h0);

    // ---- conv1: hW1 = h0 @ W1 ; h1 = scatter(hW1) + hW1*dinv^2 + b1
    mm_k9_kernel   <<<NV / T, T, 0, stream>>>(h0, W1, bufA);
    self_kernel    <<<NV / T, T, 0, stream>>>(bufA, dinv, b1, bufB);
    scatter_kernel <<<EV / T, T, 0, stream>>>(ei, bufA, dinv, bufB);

    // ---- conv2: hW2 = relu(h1) @ W2 (WMMA, relu fused into A-tile load)
    gemm_wmma_kernel<<<N_NODES / 16, T, 0, stream>>>(bufB, W2, nullptr, bufA,
                                                     N_NODES, /*applyRelu=*/1);
    self_kernel    <<<NV / T, T, 0, stream>>>(bufA, dinv, b2, bufB);
    scatter_kernel <<<EV / T, T, 0, stream>>>(ei, bufA, dinv, bufB);

    // ---- mean pool
    zero_f_kernel<<<(N_GRAPHS * HIDDEN + N_GRAPHS + T - 1) / T, T, 0, stream>>>(
        gs, N_GRAPHS * HIDDEN + N_GRAPHS);
    pool_kernel<<<NV / T, T, 0, stream>>>(batch, bufB, gs, cnt);
    mean_kernel<<<(N_GRAPHS * HIDDEN) / T, T, 0, stream>>>(gs, cnt);

    // ---- final FC: out = graph_emb @ Wfc + bfc (WMMA)
    gemm_wmma_kernel<<<N_GRAPHS / 16, T, 0, stream>>>(gs, Wfc, bfc, out,
                                                      N_GRAPHS, /*applyRelu=*/0);
}